// Encoder_90469191122997
// MI455X (gfx1250) — compile-verified
//
#include <hip/hip_runtime.h>

// Problem constants (match reference)
constexpr int Tn = 6, Bn = 8, Cn = 192, Hn = 112, Wn = 112;
constexpr int Kn = 3, H0 = 56, W0 = 56;
constexpr int Nn = Tn * Kn;            // 18 graph nodes
constexpr int HW = Hn * Wn;            // 12544 pixels / plane
constexpr int MSZ = H0 * W0;           // 3136 mask cells
constexpr int CCHUNK = 8;              // channels per streaming block
constexpr int NCC = Cn / CCHUNK;       // 24 chunks

typedef float v2f __attribute__((ext_vector_type(2)));
typedef float v8f __attribute__((ext_vector_type(8)));

// ---------------------------------------------------------------------------
// Kernel 1: node[b, t*K+k, c] = (1/HW) * sum_{h,w} x[t,b,c,h,w] * mask_k[h/2,w/2]
// One block per (t, b, channel-chunk). Mask bits packed into LDS once.
// ---------------------------------------------------------------------------
__global__ void __launch_bounds__(256) k_node(const float* __restrict__ x,
                                              const int* __restrict__ gmask,
                                              float* __restrict__ node) {
  __shared__ int smask[MSZ];
  __shared__ float sacc[Kn];
  const int tid = threadIdx.x;
  const int bid = blockIdx.x;
  const int cc  = bid % NCC;
  const int tb  = bid / NCC;
  const int b   = tb % Bn;
  const int t   = tb / Bn;

  const int mbase = ((b * Tn + t) * Kn) * MSZ;
  for (int i = tid; i < MSZ; i += 256) {
    int m0 = gmask[mbase + i] & 1;
    int m1 = gmask[mbase + MSZ + i] & 1;
    int m2 = gmask[mbase + 2 * MSZ + i] & 1;
    smask[i] = m0 | (m1 << 1) | (m2 << 2);
  }
  __syncthreads();

  for (int ci = 0; ci < CCHUNK; ++ci) {
    const int c = cc * CCHUNK + ci;
    const float* xp = x + (((size_t)t * Bn + b) * Cn + c) * (size_t)HW;
    if (tid < Kn) sacc[tid] = 0.f;
    __syncthreads();

    float a0 = 0.f, a1 = 0.f, a2 = 0.f;
    for (int p = tid; p < HW; p += 256) {
      __builtin_prefetch(xp + p + 1024, 0, 0);   // global_prefetch_b8, speculative
      float v = xp[p];
      int h = p / Wn, w = p - h * Wn;
      int mm = smask[(h >> 1) * W0 + (w >> 1)];
      if (mm & 1) a0 += v;
      if (mm & 2) a1 += v;
      if (mm & 4) a2 += v;
    }
#pragma unroll
    for (int off = 16; off > 0; off >>= 1) {     // wave32 reduction
      a0 += __shfl_down(a0, off, 32);
      a1 += __shfl_down(a1, off, 32);
      a2 += __shfl_down(a2, off, 32);
    }
    if ((tid & 31) == 0) {
      atomicAdd(&sacc[0], a0);                   // ds_add_f32
      atomicAdd(&sacc[1], a1);
      atomicAdd(&sacc[2], a2);
    }
    __syncthreads();
    if (tid < Kn)
      node[((size_t)b * Nn + (t * Kn + tid)) * Cn + c] = sacc[tid] * (1.f / (float)HW);
    __syncthreads();
  }
}

// ---------------------------------------------------------------------------
// Kernel 2: per-batch graph math, full f32.  One block (8 waves) per batch.
//   S = node@node^T -> softmax -> adj     (tiny, VALU)
//   aaa = node @ W_emb^T                   (WMMA f32 16x16x4)
//   support = aaa @ W_gcn                  (WMMA f32 16x16x4)
//   out = adj @ support + b_gcn            (tiny, VALU)
// sA holds node (padded 32 rows), then is reused for support.
// ---------------------------------------------------------------------------
__global__ void __launch_bounds__(256) k_gcn(const float* __restrict__ node,
                                             const float* __restrict__ W_emb,
                                             const float* __restrict__ W_gcn,
                                             const float* __restrict__ b_gcn,
                                             float* __restrict__ outg) {
  __shared__ float sA[32 * Cn];   // node rows 0..17, zero-padded to 32; later: support
  __shared__ float sB[32 * Cn];   // aaa
  __shared__ float sS[Nn * Nn];   // scores -> adj

  const int tid  = threadIdx.x;
  const int lane = tid & 31;
  const int wave = tid >> 5;
  const int b    = blockIdx.x;

  for (int i = tid; i < 32 * Cn; i += 256) {
    int n = i / Cn;
    sA[i] = (n < Nn) ? node[(size_t)b * Nn * Cn + i] : 0.f;
  }
  __syncthreads();

  // S = node @ node^T  (18x18)
  for (int e = tid; e < Nn * Nn; e += 256) {
    int n = e / Nn, m = e % Nn;
    float s = 0.f;
    for (int c = 0; c < Cn; ++c) s += sA[n * Cn + c] * sA[m * Cn + c];
    sS[e] = s;
  }
  __syncthreads();

  // row softmax -> adj (threads 0..17 own one row each)
  if (tid < Nn) {
    float mx = -1e30f;
    for (int m = 0; m < Nn; ++m) mx = fmaxf(mx, sS[tid * Nn + m]);
    float sum = 0.f;
    for (int m = 0; m < Nn; ++m) {
      float ev = __expf(sS[tid * Nn + m] - mx);
      sS[tid * Nn + m] = ev;
      sum += ev;
    }
    float inv = 1.f / sum;
    for (int m = 0; m < Nn; ++m) sS[tid * Nn + m] *= inv;
  }
  __syncthreads();

  const int r  = lane & 15;           // row/col within tile
  const int hi = lane >> 4;           // lane half
  const int kh = hi << 1;             // K sub-offset {0,2}

  // aaa = node @ W_emb^T   -> sB.   D[n][d] = sum_c A[n][c] * W_emb[d][c]
  for (int tile = wave; tile < 2 * (Cn / 16); tile += 8) {   // 24 tiles, uniform per wave
    const int mt  = tile & 1;
    const int dt  = tile >> 1;
    const int row = mt * 16 + r;
    const int col = dt * 16 + r;
    v8f acc = {0.f, 0.f, 0.f, 0.f, 0.f, 0.f, 0.f, 0.f};
    for (int k0 = 0; k0 < Cn; k0 += 4) {
      v2f a, bb;
      a[0]  = sA[row * Cn + k0 + kh];
      a[1]  = sA[row * Cn + k0 + kh + 1];
      bb[0] = W_emb[(size_t)col * Cn + k0 + kh];       // B[K][d] = W_emb[d][K]
      bb[1] = W_emb[(size_t)col * Cn + k0 + kh + 1];
      acc = __builtin_amdgcn_wmma_f32_16x16x4_f32(false, a, false, bb,
                                                  (short)0, acc, false, false);
    }
#pragma unroll
    for (int v = 0; v < 8; ++v) {
      int m = mt * 16 + (hi << 3) + v;
      sB[m * Cn + dt * 16 + r] = acc[v];
    }
  }
  __syncthreads();

  // support = aaa @ W_gcn   -> sA (node no longer needed)
  for (int tile = wave; tile < 2 * (Cn / 16); tile += 8) {
    const int mt  = tile & 1;
    const int et  = tile >> 1;
    const int row = mt * 16 + r;
    const int col = et * 16 + r;
    v8f acc = {0.f, 0.f, 0.f, 0.f, 0.f, 0.f, 0.f, 0.f};
    for (int k0 = 0; k0 < Cn; k0 += 4) {
      v2f a, bb;
      a[0]  = sB[row * Cn + k0 + kh];
      a[1]  = sB[row * Cn + k0 + kh + 1];
      bb[0] = W_gcn[(size_t)(k0 + kh) * Cn + col];     // B[K][e] = W_gcn[K][e]
      bb[1] = W_gcn[(size_t)(k0 + kh + 1) * Cn + col];
      acc = __builtin_amdgcn_wmma_f32_16x16x4_f32(false, a, false, bb,
                                                  (short)0, acc, false, false);
    }
#pragma unroll
    for (int v = 0; v < 8; ++v) {
      int m = mt * 16 + (hi << 3) + v;
      sA[m * Cn + et * 16 + r] = acc[v];
    }
  }
  __syncthreads();

  // out = adj @ support + b_gcn  (K=18, tiny)
  for (int i = tid; i < Nn * Cn; i += 256) {
    int n = i / Cn, e = i - n * Cn;
    float s = b_gcn[e];
    for (int m = 0; m < Nn; ++m) s += sS[n * Nn + m] * sA[m * Cn + e];
    outg[(size_t)b * Nn * Cn + i] = s;
  }
}

// ---------------------------------------------------------------------------
// Kernel 3: y[t,b,c,h,w] = x[t,b,c,h,w] + sum_k out[b,t,k,c] * mask_k[h/2,w/2]
// ---------------------------------------------------------------------------
__global__ void __launch_bounds__(256) k_resid(const float* __restrict__ x,
                                               const int* __restrict__ gmask,
                                               const float* __restrict__ outg,
                                               float* __restrict__ y) {
  __shared__ int smask[MSZ];
  const int tid = threadIdx.x;
  const int bid = blockIdx.x;
  const int cc  = bid % NCC;
  const int tb  = bid / NCC;
  const int b   = tb % Bn;
  const int t   = tb / Bn;

  const int mbase = ((b * Tn + t) * Kn) * MSZ;
  for (int i = tid; i < MSZ; i += 256) {
    int m0 = gmask[mbase + i] & 1;
    int m1 = gmask[mbase + MSZ + i] & 1;
    int m2 = gmask[mbase + 2 * MSZ + i] & 1;
    smask[i] = m0 | (m1 << 1) | (m2 << 2);
  }
  __syncthreads();

  for (int ci = 0; ci < CCHUNK; ++ci) {
    const int c = cc * CCHUNK + ci;
    const size_t plane = (((size_t)t * Bn + b) * Cn + c) * (size_t)HW;
    const float o0 = outg[((size_t)b * Nn + t * Kn + 0) * Cn + c];
    const float o1 = outg[((size_t)b * Nn + t * Kn + 1) * Cn + c];
    const float o2 = outg[((size_t)b * Nn + t * Kn + 2) * Cn + c];
    const float* xp = x + plane;
    float* yp = y + plane;
    for (int p = tid; p < HW; p += 256) {
      __builtin_prefetch(xp + p + 1024, 0, 0);
      float v = xp[p];
      int h = p / Wn, w = p - h * Wn;
      int mm = smask[(h >> 1) * W0 + (w >> 1)];
      float rr = v;
      if (mm & 1) rr += o0;
      if (mm & 2) rr += o1;
      if (mm & 4) rr += o2;
      yp[p] = rr;
    }
  }
}

// ---------------------------------------------------------------------------
extern "C" void kernel_launch(void* const* d_in, const int* in_sizes, int n_in,
                              void* d_out, int out_size, void* d_ws, size_t ws_size,
                              hipStream_t stream) {
  const float* x     = (const float*)d_in[0];
  const int*   masks = (const int*)d_in[1];
  const float* W_emb = (const float*)d_in[2];
  const float* W_gcn = (const float*)d_in[3];
  const float* b_gcn = (const float*)d_in[4];
  float* y = (float*)d_out;

  float* node = (float*)d_ws;                 // [B, 18, 192]
  float* outg = node + (size_t)Bn * Nn * Cn;  // [B, 18, 192]

  dim3 gridS(Tn * Bn * NCC);                  // 1152 streaming blocks
  k_node <<<gridS,   256, 0, stream>>>(x, masks, node);
  k_gcn  <<<dim3(Bn), 256, 0, stream>>>(node, W_emb, W_gcn, b_gcn, outg);
  k_resid<<<gridS,   256, 0, stream>>>(x, masks, outg, y);
}